// DP_2911987826885
// MI455X (gfx1250) — compile-verified
//
#include <hip/hip_runtime.h>
#include <hip/hip_bf16.h>

typedef __attribute__((ext_vector_type(2))) float v2f;
typedef __attribute__((ext_vector_type(8))) float v8f;

#define NTYPES 2
#define BB     4
#define NN     512
#define MM     200      // neighbors per atom (MNEI*NTYPES)
#define MH     100      // neighbors per type block
#define EMB    100
#define FITH   240
#define DDIM   1600     // EMB * M2

// ---------------------------------------------------------------------------
// Kernel 1: per-atom descriptor.  One workgroup (128 threads) per atom.
//   Rn (smooth cutoff, normalized) -> embedding MLP per neighbor ->
//   xyz = Rn^T @ G / 200 -> D[c*16+d] = sum_k xyz[k][c]*xyz[k][d]
// ---------------------------------------------------------------------------
__global__ __launch_bounds__(128) void dp_descriptor(
    const int*   __restrict__ typemap,   // [N]
    const float* __restrict__ imagedr,   // [B,N,200,4]
    const float* __restrict__ davg,      // [2,200,4]
    const float* __restrict__ dstd,      // [2,200,4]
    const float* __restrict__ eW0, const float* __restrict__ eb0,   // (2,2,1,25),(2,2,25)
    const float* __restrict__ eW1, const float* __restrict__ eb1,   // (2,2,25,50),(2,2,50)
    const float* __restrict__ eW2, const float* __restrict__ eb2,   // (2,2,50,100),(2,2,100)
    float*       __restrict__ Dws)       // [2048,1600]
{
    __shared__ float sRn[MM][4];          // 3.2 KB
    __shared__ float sH0[MH][25];         // 10 KB
    __shared__ float sH1[MH][51];         // 20.4 KB (padded)
    __shared__ float sG[MH][51];          // 20.4 KB (50-col chunk, padded)
    __shared__ float sxyz[4][EMB];        // 1.6 KB

    const int atom  = blockIdx.x;             // b*N + n
    const int n     = atom & (NN - 1);
    const int tid   = threadIdx.x;
    const int itype = typemap[n];

    // ---- Rn ----
    for (int m = tid; m < MM; m += 128) {
        const float* dr = imagedr + (size_t)(atom * MM + m) * 4;
        float R     = dr[0];
        float Rsafe = (R > 1e-5f) ? R : 1.0f;
        float u     = (R - 0.5f) * (1.0f / 5.5f);
        float mid   = (u*u*u*((-6.0f*u + 15.0f)*u - 10.0f) + 1.0f) / Rsafe;
        float S;
        if      (R > 0.0f && R < 0.5f) S = 1.0f / Rsafe;
        else if (R > 0.5f && R < 6.0f) S = mid;
        else                           S = 0.0f;
        float SR  = S / Rsafe;
        bool  msk = fabsf(R) > 1e-5f;
        float Ri0 = S;
        float Ri1 = msk ? SR * dr[1] : 0.0f;
        float Ri2 = msk ? SR * dr[2] : 0.0f;
        float Ri3 = msk ? SR * dr[3] : 0.0f;
        const float* av = davg + (size_t)(itype * MM + m) * 4;
        const float* sv = dstd + (size_t)(itype * MM + m) * 4;
        sRn[m][0] = (Ri0 - av[0]) / sv[0];
        sRn[m][1] = (Ri1 - av[1]) / sv[1];
        sRn[m][2] = (Ri2 - av[2]) / sv[2];
        sRn[m][3] = (Ri3 - av[3]) / sv[3];
    }
    if (tid < EMB) {
        sxyz[0][tid] = 0.0f; sxyz[1][tid] = 0.0f;
        sxyz[2][tid] = 0.0f; sxyz[3][tid] = 0.0f;
    }
    __syncthreads();

    for (int j = 0; j < NTYPES; ++j) {
        const int ij = itype * NTYPES + j;
        const float* W0 = eW0 + ij * 25;
        const float* b0 = eb0 + ij * 25;
        const float* W1 = eW1 + ij * (25 * 50);
        const float* b1 = eb1 + ij * 50;
        const float* W2 = eW2 + ij * (50 * 100);
        const float* b2 = eb2 + ij * 100;

        // thread tid < 100 owns neighbor m = tid of this type block
        if (tid < MH) {
            float s = sRn[j * MH + tid][0];
            for (int c = 0; c < 25; ++c)
                sH0[tid][c] = tanhf(s * W0[c] + b0[c]);
            for (int c = 0; c < 50; ++c) {
                float acc = b1[c];
                for (int k = 0; k < 25; ++k) acc += sH0[tid][k] * W1[k * 50 + c];
                sH1[tid][c] = tanhf(acc) + sH0[tid][c < 25 ? c : c - 25];
            }
        }
        __syncthreads();

        // G in two 50-column chunks; accumulate xyz per chunk
        for (int chunk = 0; chunk < 2; ++chunk) {
            const int c0 = chunk * 50;
            if (tid < MH) {
                for (int c = 0; c < 50; ++c) {
                    float acc = b2[c0 + c];
                    for (int k = 0; k < 50; ++k) acc += sH1[tid][k] * W2[k * EMB + c0 + c];
                    sG[tid][c] = tanhf(acc) + sH1[tid][c];   // (c0+c)%50 == c
                }
            }
            __syncthreads();
            if (tid < 50) {
                float xa0 = 0.f, xa1 = 0.f, xa2 = 0.f, xa3 = 0.f;
                for (int m = 0; m < MH; ++m) {
                    float g  = sG[m][tid];
                    float r0 = sRn[j * MH + m][0];
                    float r1 = sRn[j * MH + m][1];
                    float r2 = sRn[j * MH + m][2];
                    float r3 = sRn[j * MH + m][3];
                    xa0 += r0 * g; xa1 += r1 * g; xa2 += r2 * g; xa3 += r3 * g;
                }
                sxyz[0][c0 + tid] += xa0;
                sxyz[1][c0 + tid] += xa1;
                sxyz[2][c0 + tid] += xa2;
                sxyz[3][c0 + tid] += xa3;
            }
            __syncthreads();
        }
    }

    if (tid < EMB) {
        const float inv = 1.0f / (float)MM;
        sxyz[0][tid] *= inv; sxyz[1][tid] *= inv;
        sxyz[2][tid] *= inv; sxyz[3][tid] *= inv;
    }
    __syncthreads();

    for (int idx = tid; idx < DDIM; idx += 128) {
        int c = idx >> 4, d = idx & 15;
        float v = sxyz[0][c] * sxyz[0][d] + sxyz[1][c] * sxyz[1][d]
                + sxyz[2][c] * sxyz[2][d] + sxyz[3][c] * sxyz[3][d];
        Dws[(size_t)atom * DDIM + idx] = v;
    }
}

// ---------------------------------------------------------------------------
// Kernel 2: fitting net with f32 WMMA + async global->LDS A staging.
// One workgroup (8 waves / 256 thr) per 16 atoms.  Each wave owns two
// 16-column tiles of the 240-wide layer.
// ISA f32 16x16x4 operand layout (wave32):
//   A (16x4):  lane = half*16+r  -> a.{x,y} = A[r][2*half + {0,1}]
//   B (4x16):  lane = half*16+r  -> b.{x,y} = B[2*half + {0,1}][r]
//   C/D 16x16: VGPR v at lane    -> D[v + 8*half][r]
// Layer-0 A tile (16x1600) is streamed through LDS in 25 chunks of K=64
// (exactly one global_load_async_to_lds_b128 per thread per chunk ->
// per-wave ASYNCcnt is deterministic), double-buffered so the async engine
// copies chunk ch+1 while WMMAs consume chunk ch.
// ---------------------------------------------------------------------------
#define CK   64                 // K-chunk (floats per row)
#define NCH  (DDIM / CK)        // 25 chunks
#define ARS  68                 // A-chunk LDS row stride: 272B, 16B aligned

__global__ __launch_bounds__(256) void dp_fit(
    const int*   __restrict__ typemap,  // [N]
    const float* __restrict__ Dws,      // [2048,1600]
    const float* __restrict__ fW0, const float* __restrict__ fb0,  // (2,1600,240),(2,240)
    const float* __restrict__ fW1, const float* __restrict__ fb1,  // (2,240,240),(2,240)
    const float* __restrict__ fW2, const float* __restrict__ fb2,  // (2,240,240),(2,240)
    const float* __restrict__ fW3, const float* __restrict__ fb3,  // (2,240,1),(2,1)
    const float* __restrict__ eshift,   // [2]
    float*       __restrict__ Ei)       // [2048]  (= d_out + B)
{
    __shared__ float sH [16][FITH + 1];
    __shared__ float sH2[16][FITH + 1];
    __shared__ __align__(16) float sA[2][16][ARS];   // 8.7 KB double buffer
    __shared__ float sE [NTYPES][16];

    const int tid     = threadIdx.x;
    const int wave    = tid >> 5;
    const int lane    = tid & 31;
    const int half    = lane >> 4;
    const int r       = lane & 15;
    const int rowbase = blockIdx.x * 16;

    // this wave's two column tiles (tile 15 does not exist: clamp, discard acc1)
    const int  t1    = wave + 8;
    const bool has1  = (t1 < 15);
    const int  col0  = wave * 16 + r;
    const int  col1  = has1 ? (t1 * 16 + r) : col0;

    // async A-chunk issue: one B128 (4 floats) per thread; 256 thr = 16 rows x 16 segs
    const int arow = tid >> 4;        // 0..15
    const int aseg = tid & 15;        // 0..15
    const float* gArow = Dws + (size_t)(rowbase + arow) * DDIM + aseg * 4;

    for (int it = 0; it < NTYPES; ++it) {
        const float* W0 = fW0 + (size_t)it * DDIM * FITH;
        const float* b0 = fb0 + it * FITH;
        const float* W1 = fW1 + it * FITH * FITH;
        const float* b1 = fb1 + it * FITH;
        const float* W2 = fW2 + it * FITH * FITH;
        const float* b2 = fb2 + it * FITH;

        // ---- layer 0: sH = tanh(D @ W0 + b0), K = 1600, A via async LDS ----
        {
            v8f acc0 = {0.f,0.f,0.f,0.f,0.f,0.f,0.f,0.f};
            v8f acc1 = {0.f,0.f,0.f,0.f,0.f,0.f,0.f,0.f};

            {   // prime chunk 0
                unsigned lds = (unsigned)(size_t)&sA[0][arow][aseg * 4];
                const float* g = gArow;
                asm volatile("global_load_async_to_lds_b128 %0, %1, off"
                             :: "v"(lds), "v"(g) : "memory");
            }
            for (int ch = 0; ch < NCH; ++ch) {
                asm volatile("s_wait_asynccnt 0x0" ::: "memory");
                __syncthreads();                       // chunk ch visible everywhere
                if (ch + 1 < NCH) {                    // stream next chunk
                    unsigned lds = (unsigned)(size_t)&sA[(ch + 1) & 1][arow][aseg * 4];
                    const float* g = gArow + (ch + 1) * CK;
                    asm volatile("global_load_async_to_lds_b128 %0, %1, off"
                                 :: "v"(lds), "v"(g) : "memory");
                }
                const int buf = ch & 1;
                for (int kk = 0; kk < CK; kk += 4) {
                    const int kl = kk + 2 * half;
                    const int kg = ch * CK + kl;
                    v2f a;  a.x = sA[buf][r][kl];
                            a.y = sA[buf][r][kl + 1];
                    v2f bA; bA.x = W0[(size_t)kg * FITH + col0];
                            bA.y = W0[(size_t)(kg + 1) * FITH + col0];
                    acc0 = __builtin_amdgcn_wmma_f32_16x16x4_f32(
                               false, a, false, bA, (short)0, acc0, false, false);
                    v2f bB; bB.x = W0[(size_t)kg * FITH + col1];
                            bB.y = W0[(size_t)(kg + 1) * FITH + col1];
                    acc1 = __builtin_amdgcn_wmma_f32_16x16x4_f32(
                               false, a, false, bB, (short)0, acc1, false, false);
                }
                __syncthreads();                       // done reading buf before reuse
            }
            const float bias0 = b0[col0];
            #pragma unroll
            for (int v = 0; v < 8; ++v)
                sH[v + 8 * half][col0] = tanhf(acc0[v] + bias0);
            if (has1) {
                const float bias1 = b0[col1];
                #pragma unroll
                for (int v = 0; v < 8; ++v)
                    sH[v + 8 * half][col1] = tanhf(acc1[v] + bias1);
            }
        }
        __syncthreads();

        // ---- layer 1: sH2 = sH + tanh(sH @ W1 + b1), K = 240 ----
        for (int tile = wave; tile < 15; tile += 8) {
            const int col = tile * 16 + r;
            v8f acc = {0.f,0.f,0.f,0.f,0.f,0.f,0.f,0.f};
            for (int kb = 0; kb < FITH; kb += 4) {
                const int k0 = kb + 2 * half;
                v2f a; a.x = sH[r][k0];
                       a.y = sH[r][k0 + 1];
                v2f b; b.x = W1[k0 * FITH + col];
                       b.y = W1[(k0 + 1) * FITH + col];
                acc = __builtin_amdgcn_wmma_f32_16x16x4_f32(
                          false, a, false, b, (short)0, acc, false, false);
            }
            const float bias = b1[col];
            #pragma unroll
            for (int v = 0; v < 8; ++v) {
                const int row = v + 8 * half;
                sH2[row][col] = sH[row][col] + tanhf(acc[v] + bias);
            }
        }
        __syncthreads();

        // ---- layer 2: sH = sH2 + tanh(sH2 @ W2 + b2), K = 240 ----
        for (int tile = wave; tile < 15; tile += 8) {
            const int col = tile * 16 + r;
            v8f acc = {0.f,0.f,0.f,0.f,0.f,0.f,0.f,0.f};
            for (int kb = 0; kb < FITH; kb += 4) {
                const int k0 = kb + 2 * half;
                v2f a; a.x = sH2[r][k0];
                       a.y = sH2[r][k0 + 1];
                v2f b; b.x = W2[k0 * FITH + col];
                       b.y = W2[(k0 + 1) * FITH + col];
                acc = __builtin_amdgcn_wmma_f32_16x16x4_f32(
                          false, a, false, b, (short)0, acc, false, false);
            }
            const float bias = b2[col];
            #pragma unroll
            for (int v = 0; v < 8; ++v) {
                const int row = v + 8 * half;
                sH[row][col] = sH2[row][col] + tanhf(acc[v] + bias);
            }
        }
        __syncthreads();

        // ---- layer 3: e = sH @ W3 + b3 ----
        if (tid < 16) {
            const float* W3 = fW3 + it * FITH;
            float acc = fb3[it];
            for (int c = 0; c < FITH; ++c) acc += sH[tid][c] * W3[c];
            sE[it][tid] = acc;
        }
        __syncthreads();
    }

    if (tid < 16) {
        const int atom = rowbase + tid;
        const int t    = typemap[atom & (NN - 1)];
        Ei[atom] = sE[t][tid] + eshift[t];
    }
}

// ---------------------------------------------------------------------------
// Kernel 3: Etot[b] = sum_n Ei[b,n]
// ---------------------------------------------------------------------------
__global__ __launch_bounds__(256) void dp_etot(const float* __restrict__ Ei,
                                               float* __restrict__ Etot)
{
    __shared__ float red[256];
    const int b = blockIdx.x;
    float s = 0.f;
    for (int n = threadIdx.x; n < NN; n += 256) s += Ei[b * NN + n];
    red[threadIdx.x] = s;
    __syncthreads();
    for (int off = 128; off > 0; off >>= 1) {
        if (threadIdx.x < off) red[threadIdx.x] += red[threadIdx.x + off];
        __syncthreads();
    }
    if (threadIdx.x == 0) Etot[b] = red[0];
}

// ---------------------------------------------------------------------------
extern "C" void kernel_launch(void* const* d_in, const int* in_sizes, int n_in,
                              void* d_out, int out_size, void* d_ws, size_t ws_size,
                              hipStream_t stream)
{
    (void)in_sizes; (void)n_in; (void)out_size; (void)ws_size;
    // input order: list_neigh, Imagetype_map, type_map, ImageDR, nghost,
    // davg, dstd, emb_W0..emb_b2, fit_W0..fit_b3, energy_shift
    const int*   typemap = (const int*)  d_in[1];
    const float* imagedr = (const float*)d_in[3];
    const float* davg    = (const float*)d_in[5];
    const float* dstd    = (const float*)d_in[6];
    const float* eW0 = (const float*)d_in[7];
    const float* eb0 = (const float*)d_in[8];
    const float* eW1 = (const float*)d_in[9];
    const float* eb1 = (const float*)d_in[10];
    const float* eW2 = (const float*)d_in[11];
    const float* eb2 = (const float*)d_in[12];
    const float* fW0 = (const float*)d_in[13];
    const float* fb0 = (const float*)d_in[14];
    const float* fW1 = (const float*)d_in[15];
    const float* fb1 = (const float*)d_in[16];
    const float* fW2 = (const float*)d_in[17];
    const float* fb2 = (const float*)d_in[18];
    const float* fW3 = (const float*)d_in[19];
    const float* fb3 = (const float*)d_in[20];
    const float* esh = (const float*)d_in[21];

    float* out = (float*)d_out;          // [0..3]=Etot, [4..2051]=Ei
    float* Dws = (float*)d_ws;           // [2048][1600] f32 = 13.1 MB

    dp_descriptor<<<BB * NN, 128, 0, stream>>>(typemap, imagedr, davg, dstd,
                                               eW0, eb0, eW1, eb1, eW2, eb2, Dws);
    dp_fit<<<(BB * NN) / 16, 256, 0, stream>>>(typemap, Dws,
                                               fW0, fb0, fW1, fb1, fW2, fb2,
                                               fW3, fb3, esh, out + BB);
    dp_etot<<<BB, 256, 0, stream>>>(out + BB, out);
}